// RAVENlog_74698071212084
// MI455X (gfx1250) — compile-verified
//
#include <hip/hip_runtime.h>

// RAVENlog: memory-bound elementwise kernel for MI455X (gfx1250, wave32).
// 67.1M fp32 in -> 67.1M fp32 out: 512MB stream => HBM-bound (~22us floor
// at 23.3 TB/s). Strategy: 128-bit non-temporal global loads/stores
// (global_load_b128 / global_store_b128 with TH=NT), exponent extraction via
// integer bit ops (exact floor(log2 x) for positive normals), v_rndne_f32 for
// round-half-to-even quantization, v_ldexp_f32 for exact 2^k scale factors.

typedef float v4f __attribute__((ext_vector_type(4)));

#define LN2F   0.6931471805599453f
#define POINTF 0.75f

struct QConsts {
    float c[4];    // Taylor coefficients
    float s[4];    // 2^fracwidth
    float inv[4];  // 2^-fracwidth (exact)
    float lo[4];   // -2^intwidth
    float hi[4];   //  2^intwidth - 2^-fracwidth
    float q3;      // quantize(coeff[3]) -- input-independent first Horner step
};

__device__ __forceinline__ float quantize1(float v, float s, float inv,
                                           float lo, float hi) {
    // round-half-to-even on the 2^-fw grid, then clamp to signed range
    float q = __builtin_rintf(v * s) * inv;   // v_rndne_f32 (RNE mode default)
    return fminf(fmaxf(q, lo), hi);
}

__device__ __forceinline__ float raven_eval(float xv, const QConsts& qc) {
    // x is strictly positive and normal (x in [0.25, 8)):
    // floor(log2 x) == biased exponent - 127; shift = that + 1.
    unsigned bits  = __float_as_uint(xv);
    int      e     = (int)((bits >> 23) & 0xFFu) - 127;
    float    shift = (float)(e + 1);
    // x * 2^-shift == mantissa scaled into [0.5, 1): force exponent field to 126
    float mhalf = __uint_as_float((bits & 0x007FFFFFu) | 0x3F000000u);
    float var   = POINTF - mhalf;

    float out = qc.q3;                        // idx = 3 step (out was 0)
    #pragma unroll
    for (int idx = 2; idx >= 0; --idx) {
        // keep mul + add separate (XLA does not auto-fuse to fma)
        float t = out * var + qc.c[idx];
        out = quantize1(t, qc.s[idx], qc.inv[idx], qc.lo[idx], qc.hi[idx]);
    }
    return shift * LN2F - out;                // offset - poly  (scale = -1)
}

__global__ __launch_bounds__(256)
void RAVENlog_kernel(const float* __restrict__ x,
                     const float* __restrict__ coeff,
                     const float* __restrict__ intw,
                     const float* __restrict__ fracw,
                     float* __restrict__ out,
                     int n4, int n) {
    // Hoisted per-launch constants (uniform; scalarized by the compiler).
    QConsts qc;
    #pragma unroll
    for (int i = 0; i < 4; ++i) {
        qc.c[i]   = coeff[i];
        int fw    = (int)fracw[i];            // [8,7,6,5]
        int iw    = (int)intw[i];             // 7
        qc.s[i]   = __builtin_ldexpf(1.0f,  fw);   // exact 2^fw  (v_ldexp_f32)
        qc.inv[i] = __builtin_ldexpf(1.0f, -fw);   // exact 2^-fw
        float lim = __builtin_ldexpf(1.0f,  iw);
        qc.lo[i]  = -lim;
        qc.hi[i]  = lim - qc.inv[i];
    }
    qc.q3 = quantize1(qc.c[3], qc.s[3], qc.inv[3], qc.lo[3], qc.hi[3]);

    int tid = blockIdx.x * blockDim.x + threadIdx.x;
    if (tid < n4) {
        // Streaming, touch-once data: non-temporal 128-bit load/store so the
        // 512MB stream does not thrash the 192MB L2.
        const v4f* __restrict__ xv = (const v4f*)x;
        v4f vin = __builtin_nontemporal_load(xv + tid);   // global_load_b128, TH=NT
        v4f vout;
        #pragma unroll
        for (int i = 0; i < 4; ++i)
            vout[i] = raven_eval(vin[i], qc);
        __builtin_nontemporal_store(vout, ((v4f*)out) + tid); // global_store_b128, TH=NT
    }

    // Scalar tail (n not divisible by 4) -- not taken for this shape, kept for
    // determinism/robustness.
    if (blockIdx.x == 0 && threadIdx.x == 0) {
        for (int k = n4 * 4; k < n; ++k)
            out[k] = raven_eval(x[k], qc);
    }
}

extern "C" void kernel_launch(void* const* d_in, const int* in_sizes, int n_in,
                              void* d_out, int out_size, void* d_ws, size_t ws_size,
                              hipStream_t stream) {
    const float* x     = (const float*)d_in[0];
    const float* coeff = (const float*)d_in[1];
    const float* intw  = (const float*)d_in[2];
    const float* fracw = (const float*)d_in[3];
    float*       out   = (float*)d_out;

    int n  = in_sizes[0];          // 16*1024*4096 = 67,108,864
    int n4 = n >> 2;               // 4 elements per lane (128-bit accesses)
    int blocks = (n4 + 255) / 256; // 65,536 blocks of 8 wave32 each
    if (blocks < 1) blocks = 1;

    RAVENlog_kernel<<<blocks, 256, 0, stream>>>(x, coeff, intw, fracw, out, n4, n);
}